// OneDilate_84250078478479
// MI455X (gfx1250) — compile-verified
//
#include <hip/hip_runtime.h>
#include <stdint.h>

#define TILE_W 64
#define TILE_H 32
#define KSZ    10
#define PAD    4
#define IN_H   1024
#define IN_W   1024
#define OUT_H  1023
#define OUT_W  1023
#define XROWS  (TILE_H + KSZ - 1)   // 41
#define XCOLS  (TILE_W + KSZ - 1)   // 73
#define HPITCH 65                   // 64 cols + 1 pad (bank-conflict avoidance)

typedef __attribute__((ext_vector_type(2))) float v2f;
typedef __attribute__((ext_vector_type(8))) float v8f;

__device__ __forceinline__ int clampi(int v, int lo, int hi) {
    return v < lo ? lo : (v > hi ? hi : v);
}

__global__ __launch_bounds__(256)
void OneDilate_box_wmma_kernel(const float* __restrict__ x, float* __restrict__ out) {
    __shared__ float Xs[XROWS * XCOLS];   // raw input tile (clamped/replicated halo)
    __shared__ float Hs[XROWS * HPITCH];  // horizontal 10-tap sums of raw x

    const int tid = threadIdx.x;
    const int z   = blockIdx.z;               // image*channel plane, 0..47
    const int ti  = blockIdx.y * TILE_H;      // output tile row origin
    const int tj  = blockIdx.x * TILE_W;      // output tile col origin

    // --- Stage 1: async global -> LDS of the clamped input tile (raw x) ---
    const unsigned long long src =
        (unsigned long long)(uintptr_t)x + (unsigned long long)z * (IN_H * (unsigned long long)IN_W * 4ull);

    for (int idx = tid; idx < XROWS * XCOLS; idx += 256) {
        int r = idx / XCOLS;
        int c = idx - r * XCOLS;
        int grow = clampi(ti - PAD + r, 0, IN_H - 1);   // edge-replicate pad
        int gcol = clampi(tj - PAD + c, 0, IN_W - 1);
        unsigned int voff = (unsigned int)((grow * IN_W + gcol) * 4);
        unsigned int lds  = (unsigned int)(uintptr_t)(&Xs[idx]);  // low 32 bits = LDS byte offset
        asm volatile("global_load_async_to_lds_b32 %0, %1, %2"
                     :
                     : "v"(lds), "v"(voff), "s"(src)
                     : "memory");
    }
    asm volatile("s_wait_asynccnt 0x0" ::: "memory");
    __syncthreads();

    // --- Stage 2: horizontal 10-tap sums of raw x (VALU in LDS) ---
    for (int idx = tid; idx < XROWS * TILE_W; idx += 256) {
        int r = idx >> 6;         // /64
        int c = idx & 63;
        const float* row = &Xs[r * XCOLS + c];
        float s = 0.0f;
#pragma unroll
        for (int k = 0; k < KSZ; ++k) s += row[k];
        Hs[r * HPITCH + c] = s;
    }
    __syncthreads();

    // --- Stage 3: vertical 10-tap sum as a banded WMMA matmul ---
    // out = 50 - 0.5 * (100-sample box sum of x):
    //   C init = 50, A band value = -0.5, B = H rows.
    const int lane = tid & 31;
    const int wave = tid >> 5;
    const int m    = lane & 15;   // A-row / B-col within 16x16 tile
    const int half = lane >> 4;   // lane-half select (K = j + 2*half)
    const int row0 = (wave >> 2) * 16;  // 0 or 16  (2 subtile rows)
    const int col0 = (wave & 3) * 16;   // 0..48    (4 subtile cols)

    v8f acc = {50.0f, 50.0f, 50.0f, 50.0f, 50.0f, 50.0f, 50.0f, 50.0f};
#pragma unroll
    for (int k0 = 0; k0 < 28; k0 += 4) {          // K = 25 band, padded to 28
        int kA0 = k0 + 2 * half;                  // VGPR j=0 -> K = k0 + 2*half
        int kA1 = kA0 + 1;                        // VGPR j=1 -> K = k0 + 1 + 2*half
        v2f a, b;
        a.x = (kA0 >= m && kA0 <= m + 9) ? -0.5f : 0.0f;   // band of -0.5
        a.y = (kA1 >= m && kA1 <= m + 9) ? -0.5f : 0.0f;
        int hr0 = row0 + kA0; hr0 = hr0 > (XROWS - 1) ? (XROWS - 1) : hr0;  // keep B finite
        int hr1 = row0 + kA1; hr1 = hr1 > (XROWS - 1) ? (XROWS - 1) : hr1;  // (A is 0 there)
        b.x = Hs[hr0 * HPITCH + col0 + m];
        b.y = Hs[hr1 * HPITCH + col0 + m];
        acc = __builtin_amdgcn_wmma_f32_16x16x4_f32(
            /*neg_a=*/false, a, /*neg_b=*/false, b,
            /*c_mod=*/(short)0, acc, /*reuse_a=*/false, /*reuse_b=*/false);
    }

    // --- Stage 4: store D (C/D layout: M = v + 8*half, N = lane&15) ---
    const int ocol = tj + col0 + m;
    if (ocol < OUT_W) {
        float* plane = out + (size_t)z * ((size_t)OUT_H * OUT_W);
#pragma unroll
        for (int v = 0; v < 8; ++v) {
            int orow = ti + row0 + v + 8 * half;
            if (orow < OUT_H) plane[(size_t)orow * OUT_W + ocol] = acc[v];
        }
    }
}

extern "C" void kernel_launch(void* const* d_in, const int* in_sizes, int n_in,
                              void* d_out, int out_size, void* d_ws, size_t ws_size,
                              hipStream_t stream) {
    const float* x = (const float*)d_in[0];   // (16,3,1024,1024) f32
    // d_in[1] = weight: all-ones depthwise 10x10 kernel, folded into the box filter
    float* out = (float*)d_out;               // (16,3,1023,1023) f32

    dim3 grid((OUT_W + TILE_W - 1) / TILE_W,   // 16
              (OUT_H + TILE_H - 1) / TILE_H,   // 32
              16 * 3);                          // 48 planes
    OneDilate_box_wmma_kernel<<<grid, 256, 0, stream>>>(x, out);

    (void)in_sizes; (void)n_in; (void)out_size; (void)d_ws; (void)ws_size;
}